// HierarchicalGNN_87771951661224
// MI455X (gfx1250) — compile-verified
//
#include <hip/hip_runtime.h>

typedef __attribute__((ext_vector_type(2))) float v2f;
typedef __attribute__((ext_vector_type(8))) float v8f;

#define HID 128
#define NGRAPHS 64

// ---------------- degree / normalization ----------------
__global__ void deg_init(float* deg, int n) {
  int i = blockIdx.x * blockDim.x + threadIdx.x;
  if (i < n) deg[i] = 1.0f;  // self-loop contribution
}

__global__ void deg_accum(const int* __restrict__ dst, float* deg, int ne) {
  int e = blockIdx.x * blockDim.x + threadIdx.x;
  if (e < ne) atomicAdd(&deg[dst[e]], 1.0f);
}

__global__ void deg_rsqrt(float* deg, int n) {
  int i = blockIdx.x * blockDim.x + threadIdx.x;
  if (i < n) deg[i] = rsqrtf(fmaxf(deg[i], 1.0f));  // deg -> dinv in place
}

// ---------------- fp32 WMMA GEMM: C[M x 128] = A[M x 128] @ W[128 x 128] ----------------
// One wave per 16x16 tile; block = 8 waves = one 16-row stripe across all 128 cols.
__global__ __launch_bounds__(256) void gemm_wmma_f32(const float* __restrict__ A,
                                                     const float* __restrict__ W,
                                                     float* __restrict__ C) {
  const int wave = threadIdx.x >> 5;   // column tile 0..7
  const int lane = threadIdx.x & 31;
  const int half = lane >> 4;          // K sub-pair selector
  const int lr   = lane & 15;          // M (for A) / N (for B,C) within tile
  const int m0 = blockIdx.x * 16;
  const int n0 = wave * 16;

  const float* arow = A + (size_t)(m0 + lr) * HID;   // A row for this lane
  const float* bcol = W + n0 + lr;                   // W column for this lane

  v8f acc = {};
#pragma unroll
  for (int kk = 0; kk < 32; ++kk) {
    const int kb = kk * 4 + half * 2;
    v2f a, b;
    a.x = arow[kb];
    a.y = arow[kb + 1];
    b.x = bcol[(size_t)kb * HID];
    b.y = bcol[(size_t)(kb + 1) * HID];
    acc = __builtin_amdgcn_wmma_f32_16x16x4_f32(false, a, false, b,
                                                (short)0, acc, false, false);
  }
#pragma unroll
  for (int r = 0; r < 8; ++r) {
    const int m = m0 + r + 8 * half;   // C layout: VGPR r -> M=r (lanes 0-15), M=r+8 (16-31)
    C[(size_t)m * HID + n0 + lr] = acc[r];
  }
}

// ---------------- self-loop init: agg[i,:] = hw[i,:] * dinv[i]^2 ----------------
__global__ __launch_bounds__(256) void agg_init(const float* __restrict__ hw,
                                                const float* __restrict__ dinv,
                                                float* __restrict__ agg, int n) {
  int t = blockIdx.x * blockDim.x + threadIdx.x;  // over n * (HID/4) float4 groups
  if (t >= n * (HID / 4)) return;
  int node = t / (HID / 4);
  float d = dinv[node];
  float c = d * d;
  float4 v = ((const float4*)hw)[t];
  float4 o = make_float4(v.x * c, v.y * c, v.z * c, v.w * c);
  ((float4*)agg)[t] = o;
}

// ---------------- edge scatter: agg[dst] += hw[src] * dinv[src]*dinv[dst] ----------------
// One wave per edge; each lane handles 4 consecutive channels.
__global__ __launch_bounds__(256) void edge_scatter(const float* __restrict__ hw,
                                                    const float* __restrict__ dinv,
                                                    const int* __restrict__ src,
                                                    const int* __restrict__ dst,
                                                    float* __restrict__ agg, int ne) {
  unsigned gid = blockIdx.x * blockDim.x + threadIdx.x;
  unsigned e = gid >> 5;
  if (e >= (unsigned)ne) return;
  int lane = gid & 31;
  int s = src[e];
  int d = dst[e];
  float coef = dinv[s] * dinv[d];
  int c = lane * 4;
  float4 v = *(const float4*)(hw + (size_t)s * HID + c);
  float* o = agg + (size_t)d * HID + c;
  atomicAdd(o + 0, v.x * coef);
  atomicAdd(o + 1, v.y * coef);
  atomicAdd(o + 2, v.z * coef);
  atomicAdd(o + 3, v.w * coef);
}

// ---------------- bias + relu in place ----------------
__global__ __launch_bounds__(256) void bias_relu(float* __restrict__ agg,
                                                 const float* __restrict__ bias, int n) {
  int t = blockIdx.x * blockDim.x + threadIdx.x;
  if (t >= n * HID) return;
  float v = agg[t] + bias[t & (HID - 1)];
  agg[t] = v > 0.0f ? v : 0.0f;
}

// ---------------- pooling ----------------
__global__ void pool_zero(float* sums, float* cnts) {
  int t = blockIdx.x * blockDim.x + threadIdx.x;
  if (t < NGRAPHS * HID) sums[t] = 0.0f;
  if (t < NGRAPHS) cnts[t] = 0.0f;
}

__global__ __launch_bounds__(256) void pool_accum(const float* __restrict__ h,
                                                  const int* __restrict__ batch,
                                                  float* __restrict__ sums,
                                                  float* __restrict__ cnts, int n) {
  unsigned gid = blockIdx.x * blockDim.x + threadIdx.x;
  unsigned node = gid >> 5;
  if (node >= (unsigned)n) return;
  int lane = gid & 31;
  int g = batch[node];
  int c = lane * 4;
  float4 v = *(const float4*)(h + (size_t)node * HID + c);
  float* o = sums + g * HID + c;
  atomicAdd(o + 0, v.x);
  atomicAdd(o + 1, v.y);
  atomicAdd(o + 2, v.z);
  atomicAdd(o + 3, v.w);
  if (lane == 0) atomicAdd(&cnts[g], 1.0f);
}

// ---------------- MLP head ----------------
__global__ void mlp1(const float* __restrict__ sums, const float* __restrict__ cnts,
                     const float* __restrict__ W1, const float* __restrict__ b1,
                     float* __restrict__ hidden) {
  int g = blockIdx.x;   // 0..63
  int j = threadIdx.x;  // 0..63
  float ic = 1.0f / fmaxf(cnts[g], 1.0f);
  float acc = b1[j];
  for (int k = 0; k < HID; ++k) acc += (sums[g * HID + k] * ic) * W1[k * 64 + j];
  hidden[g * 64 + j] = acc > 0.0f ? acc : 0.0f;
}

__global__ void mlp2(const float* __restrict__ hidden, const float* __restrict__ W2,
                     const float* __restrict__ b2, float* __restrict__ out) {
  int t = threadIdx.x;  // 0..127
  int g = t >> 1, o = t & 1;
  float acc = b2[o];
  for (int k = 0; k < 64; ++k) acc += hidden[g * 64 + k] * W2[k * 2 + o];
  out[g * 2 + o] = acc;
}

extern "C" void kernel_launch(void* const* d_in, const int* in_sizes, int n_in,
                              void* d_out, int out_size, void* d_ws, size_t ws_size,
                              hipStream_t stream) {
  const float* x    = (const float*)d_in[0];
  const int*   ei   = (const int*)d_in[1];
  const int*   batch= (const int*)d_in[2];
  const float* W0   = (const float*)d_in[3];
  const float* b0   = (const float*)d_in[4];
  const float* Ws   = (const float*)d_in[5];
  const float* bs   = (const float*)d_in[6];
  const float* W1   = (const float*)d_in[7];
  const float* b1   = (const float*)d_in[8];
  const float* W2   = (const float*)d_in[9];
  const float* b2   = (const float*)d_in[10];
  float* out = (float*)d_out;

  const int n  = in_sizes[0] / HID;   // 50000 (multiple of 16)
  const int ne = in_sizes[1] / 2;     // 800000
  const int* esrc = ei;
  const int* edst = ei + ne;

  float* ws     = (float*)d_ws;
  float* bufA   = ws;                               // [n x 128] activations
  float* bufB   = bufA + (size_t)n * HID;           // [n x 128] h@W
  float* dinv   = bufB + (size_t)n * HID;           // [n]
  float* sums   = dinv + n;                         // [64 x 128]
  float* cnts   = sums + NGRAPHS * HID;             // [64]
  float* hidden = cnts + NGRAPHS;                   // [64 x 64]

  const int B = 256;
  // degrees -> dinv
  deg_init<<<(n + B - 1) / B, B, 0, stream>>>(dinv, n);
  deg_accum<<<(ne + B - 1) / B, B, 0, stream>>>(edst, dinv, ne);
  deg_rsqrt<<<(n + B - 1) / B, B, 0, stream>>>(dinv, n);

  const int gemm_grid    = n / 16;
  const int agg_grid     = (n * (HID / 4) + B - 1) / B;
  const int scatter_grid = ((ne * 32) + B - 1) / B;
  const int relu_grid    = (n * HID + B - 1) / B;

  // layer 0: x -> bufA
  gemm_wmma_f32<<<gemm_grid, B, 0, stream>>>(x, W0, bufB);
  agg_init<<<agg_grid, B, 0, stream>>>(bufB, dinv, bufA, n);
  edge_scatter<<<scatter_grid, B, 0, stream>>>(bufB, dinv, esrc, edst, bufA, ne);
  bias_relu<<<relu_grid, B, 0, stream>>>(bufA, b0, n);

  // layers 1..3: bufA -> bufA
  for (int l = 0; l < 3; ++l) {
    gemm_wmma_f32<<<gemm_grid, B, 0, stream>>>(bufA, Ws + (size_t)l * HID * HID, bufB);
    agg_init<<<agg_grid, B, 0, stream>>>(bufB, dinv, bufA, n);
    edge_scatter<<<scatter_grid, B, 0, stream>>>(bufB, dinv, esrc, edst, bufA, ne);
    bias_relu<<<relu_grid, B, 0, stream>>>(bufA, bs + (size_t)l * HID, n);
  }

  // global mean pool + MLP head
  pool_zero<<<(NGRAPHS * HID + B - 1) / B, B, 0, stream>>>(sums, cnts);
  pool_accum<<<((n * 32) + B - 1) / B, B, 0, stream>>>(bufA, batch, sums, cnts, n);
  mlp1<<<NGRAPHS, 64, 0, stream>>>(sums, cnts, W1, b1, hidden);
  mlp2<<<1, 128, 0, stream>>>(hidden, W2, b2, out);
}